// CEandSCLoss_13314398617736
// MI455X (gfx1250) — compile-verified
//
#include <hip/hip_runtime.h>
#include <hip/hip_bf16.h>
#include <math.h>

// Problem constants (match reference)
#define NROW 8192
#define NC   64
#define TEMP 0.5f
#define LMBD 0.5f
#define INV_TAU (1.0f / TEMP)
#define SENT  (-1.0e30f)   // finite "minus infinity": exp(SENT - finite) == 0

typedef __attribute__((ext_vector_type(2))) float v2f;
typedef __attribute__((ext_vector_type(8))) float v8f;

// ---------------------------------------------------------------------------
// Phase A: class sums S[c][k], class counts cnt[c], and CE contribution.
// ---------------------------------------------------------------------------
__global__ void __launch_bounds__(256) phaseA_kernel(
    const float* __restrict__ x, const int* __restrict__ tgt,
    float* __restrict__ S, int* __restrict__ cnt, float* __restrict__ out)
{
    __shared__ float sS[NC * NC];
    __shared__ int   sCnt[NC];
    __shared__ float sCE;

    for (int i = threadIdx.x; i < NC * NC; i += 256) sS[i] = 0.0f;
    if (threadIdx.x < NC) sCnt[threadIdx.x] = 0;
    if (threadIdx.x == 0) sCE = 0.0f;
    __syncthreads();

    const int row = blockIdx.x * 256 + threadIdx.x;   // grid = 32 blocks
    const float* xr = x + (size_t)row * NC;
    const int t = tgt[row];

    float mx = -INFINITY;
    #pragma unroll 8
    for (int k = 0; k < NC; ++k) mx = fmaxf(mx, xr[k]);

    float se = 0.0f;
    #pragma unroll 8
    for (int k = 0; k < NC; ++k) {
        float v = xr[k];
        se += __expf(v - mx);
        atomicAdd(&sS[t * NC + k], v);          // ds_add_f32
    }
    atomicAdd(&sCnt[t], 1);

    const float lse  = mx + __logf(se);
    const float ce_i = lse - xr[t];
    atomicAdd(&sCE, ce_i * ((1.0f - LMBD) / (float)NROW));
    __syncthreads();

    for (int i = threadIdx.x; i < NC * NC; i += 256) atomicAdd(&S[i], sS[i]);
    if (threadIdx.x < NC) atomicAdd(&cnt[threadIdx.x], sCnt[threadIdx.x]);
    if (threadIdx.x == 0) atomicAdd(out, sCE);
}

// ---------------------------------------------------------------------------
// Phase A2: per-row positive term  pv[i] = ((x_i . S_c - ||x_i||^2)/tau)/n_pos
// ---------------------------------------------------------------------------
__global__ void __launch_bounds__(256) phaseA2_kernel(
    const float* __restrict__ x, const int* __restrict__ tgt,
    const float* __restrict__ S, const int* __restrict__ cnt,
    float* __restrict__ pv, int* __restrict__ fl)
{
    const int row = blockIdx.x * 256 + threadIdx.x;
    const int t = tgt[row];
    const float* xr = x + (size_t)row * NC;
    const float* sc = S + (size_t)t * NC;

    float dot = 0.0f, nrm = 0.0f;
    #pragma unroll 8
    for (int k = 0; k < NC; ++k) {
        float v = xr[k];
        dot = fmaf(v, sc[k], dot);
        nrm = fmaf(v, v, nrm);
    }
    const int np = cnt[t] - 1;
    fl[row] = (np > 0) ? 1 : 0;
    pv[row] = (np > 0) ? ((dot - nrm) * INV_TAU) / (float)np : 0.0f;
}

// ---------------------------------------------------------------------------
// Phase B: streaming logsumexp of sim = (x x^T)/tau, diag excluded, via
// V_WMMA_F32_16X16X4_F32.  One wave owns 16 rows; 512 column tiles.
// B panel is staged into an explicit 16-register array so all 16 loads can
// be in flight simultaneously (clause + staged waits) before the WMMA chain.
// ---------------------------------------------------------------------------
__global__ void __launch_bounds__(256) sclB_kernel(
    const float* __restrict__ x,
    const float* __restrict__ pv, const int* __restrict__ fl,
    float* __restrict__ out)
{
    const int wave = threadIdx.x >> 5;          // 0..7
    const int lane = threadIdx.x & 31;
    const int lm   = lane & 15;                 // matrix index within tile
    const int lh   = lane >> 4;                 // lane half
    const int rowBase  = (blockIdx.x * 8 + wave) * 16;
    const int diagTile = __builtin_amdgcn_readfirstlane(rowBase >> 4); // SGPR
    const int dv       = lm - 8 * lh;           // slot v hit by diagonal (if 0..7)

    // A fragments: element (M=lm, K = 4*s + 2*lh + v), one float2 per K-step.
    v2f a[16];
    const float* arow = x + (size_t)(rowBase + lm) * NC;
    #pragma unroll
    for (int s = 0; s < 16; ++s)
        a[s] = *(const v2f*)(arow + s * 4 + 2 * lh);

    // online stats per C-fragment slot: row M = v + 8*lh, col residue = lm
    float m[8], ss[8];
    #pragma unroll
    for (int v = 0; v < 8; ++v) { m[v] = SENT; ss[v] = 0.0f; }

    #pragma unroll 2
    for (int ct = 0; ct < NROW / 16; ++ct) {
        const float* brow = x + (size_t)(ct * 16 + lm) * NC;

        // Stage the whole B panel first: 16 independent loads in flight.
        v2f b[16];
        #pragma unroll
        for (int s = 0; s < 16; ++s)
            b[s] = *(const v2f*)(brow + s * 4 + 2 * lh);

        v8f c = {};
        #pragma unroll
        for (int s = 0; s < 16; ++s)
            c = __builtin_amdgcn_wmma_f32_16x16x4_f32(
                    false, a[s], false, b[s], (short)0, c, false, false);

        float vals[8];
        #pragma unroll
        for (int v = 0; v < 8; ++v) vals[v] = c[v] * INV_TAU;

        if (ct == diagTile) {                   // SALU compare, scalar branch
            #pragma unroll
            for (int v = 0; v < 8; ++v)
                if (v == dv) vals[v] = SENT;    // mask self-similarity
        }

        // branchless online softmax, one exp per element:
        //   e = exp(min-max);  val>m ? s = s*e + 1 : s = s + e
        #pragma unroll
        for (int v = 0; v < 8; ++v) {
            float val = vals[v];
            float nm  = fmaxf(m[v], val);
            float e   = __expf(-fabsf(m[v] - val));
            bool  gt  = val > m[v];
            float c1  = gt ? e : 1.0f;
            float c2  = gt ? 1.0f : e;
            ss[v] = fmaf(ss[v], c1, c2);
            m[v]  = nm;
        }
    }

    // Merge the 16 column slots per row (butterfly within 16-lane half),
    // then lane lm==v finalizes row v + 8*lh.
    float acc = 0.0f;
    #pragma unroll
    for (int v = 0; v < 8; ++v) {
        float mm = m[v], sv = ss[v];
        #pragma unroll
        for (int off = 1; off < 16; off <<= 1) {
            float om = __shfl_xor(mm, off, 32);
            float os = __shfl_xor(sv, off, 32);
            float nm = fmaxf(mm, om);
            float e  = __expf(-fabsf(mm - om));
            sv = (om > mm) ? fmaf(sv, e, os) : fmaf(os, e, sv);
            mm = nm;
        }
        if (lm == v) {
            const int row = rowBase + v + 8 * lh;
            const float lse = mm + __logf(sv);
            acc = fl[row] ? LMBD * (lse - pv[row]) : 0.0f;
        }
    }

    // wave-level sum, one atomic per wave
    #pragma unroll
    for (int off = 1; off < 32; off <<= 1)
        acc += __shfl_xor(acc, off, 32);
    if (lane == 0) atomicAdd(out, acc);
}

// ---------------------------------------------------------------------------
// Launcher
// ---------------------------------------------------------------------------
extern "C" void kernel_launch(void* const* d_in, const int* in_sizes, int n_in,
                              void* d_out, int out_size, void* d_ws, size_t ws_size,
                              hipStream_t stream)
{
    const float* x   = (const float*)d_in[0];   // [8192,64] f32
    const int*   tgt = (const int*)d_in[1];     // [8192] int32
    float*       out = (float*)d_out;           // scalar f32

    char* ws = (char*)d_ws;
    float* S   = (float*)ws;                    ws += NC * NC * sizeof(float);
    int*   cnt = (int*)ws;                      ws += NC * sizeof(int);
    float* pv  = (float*)ws;                    ws += NROW * sizeof(float);
    int*   fl  = (int*)ws;

    hipMemsetAsync(out, 0, sizeof(float), stream);
    hipMemsetAsync(S, 0, (NC * NC) * sizeof(float) + NC * sizeof(int), stream);

    phaseA_kernel <<<NROW / 256, 256, 0, stream>>>(x, tgt, S, cnt, out);
    phaseA2_kernel<<<NROW / 256, 256, 0, stream>>>(x, tgt, S, cnt, pv, fl);
    sclB_kernel   <<<NROW / 128, 256, 0, stream>>>(x, pv, fl, out);
}